// MeshLoss2D_2834678415376
// MI455X (gfx1250) — compile-verified
//
#include <hip/hip_runtime.h>

// ---------------------------------------------------------------------------
// MeshLoss2D on MI455X (gfx1250):
//   d(p,m) = ||p||^2 + ||m||^2 - 2 p.m  as a K=4 homogeneous GEMM via
//   V_WMMA_F32_16X16X4_F32:  A row = [-2px,-2py,-2pz, 1],
//                            B col = [ mx,  my,  mz, ||m||^2]
//   ||p||^2 is per-row constant -> added after the min over mesh columns.
// Data movement: per-batch packed mesh tiles (141.5 KB) are DMA'd into LDS
// once per workgroup by the Tensor Data Mover (tensor_load_to_lds +
// s_wait_tensorcnt), then the WMMA loop feeds from ds_load_b64.
// ---------------------------------------------------------------------------

typedef __attribute__((ext_vector_type(2))) float v2f;
typedef __attribute__((ext_vector_type(8))) float v8f;
typedef unsigned int uint32x4 __attribute__((ext_vector_type(4)));
typedef int int32x8 __attribute__((ext_vector_type(8)));
typedef int int32x4 __attribute__((ext_vector_type(4)));

#define BATCH 4
#define CH    3
#define MH    32
#define MW    32
#define FACT  3
#define OH    94                 // (32-1)*3+1
#define OW    94
#define NPTS  (OH * OW)          // 8836 mesh points per batch
#define NT    553                // ceil(8836 / 16) mesh tiles per batch
#define MPTS  4096               // pc points per batch
#define MT    (MPTS / 16)        // 256 pc tiles per batch
#define NWAVE (BATCH * MT)       // 1024 waves total
#define TILE_BYTES 256           // one mesh tile = 32 float2 = 256 B
#define LDS_BYTES  (NT * TILE_BYTES)   // 141568 B (< 320 KB WGP LDS)
#define WS_TILES_OFF 4096        // byte offset of mesh tiles inside d_ws

// v_min_num_f32 without fminf's sNaN canonicalization (inputs are WMMA
// outputs / previous mins — never signaling NaN).
static __device__ __forceinline__ float min_fast(float a, float b) {
  float r;
  asm("v_min_num_f32 %0, %1, %2" : "=v"(r) : "v"(a), "v"(b));
  return r;
}

// ---------------------------------------------------------------------------
// Kernel 1: bilinear mesh refinement + pack into WMMA-B-fragment tiles.
// Tile t stores 32 float2; lane L reads tile[L]:
//   L in [0,16) : (k0,k1) = (mx, my)        for mesh point n = 16*t + L
//   L in [16,32): (k2,k3) = (mz, ||m||^2)   for mesh point n = 16*t + (L-16)
// Padded points (n >= 8836) get ||m||^2 = 1e30 so they never win the min.
// ---------------------------------------------------------------------------
__global__ void refine_pack_kernel(const float* __restrict__ mesh,
                                   float2* __restrict__ tiles) {
  int tid = blockIdx.x * blockDim.x + threadIdx.x;
  if (tid >= BATCH * NT * 16) return;
  int b  = tid / (NT * 16);
  int r  = tid - b * (NT * 16);
  int t  = r >> 4;
  int nl = r & 15;
  int n  = t * 16 + nl;

  float mx = 0.f, my = 0.f, mz = 0.f, s = 1e30f;
  if (n < NPTS) {
    int oy = n / OW;
    int ox = n - oy * OW;
    float ys = (float)oy / (float)FACT;     // identical arithmetic to reference
    float xs = (float)ox / (float)FACT;
    int y0 = (int)floorf(ys); y0 = (y0 > MH - 2) ? (MH - 2) : y0;
    int x0 = (int)floorf(xs); x0 = (x0 > MW - 2) ? (MW - 2) : x0;
    float wy = ys - (float)y0;
    float wx = xs - (float)x0;
    const float* mb = mesh + (size_t)b * CH * MH * MW;
    float v[3];
#pragma unroll
    for (int c = 0; c < 3; ++c) {
      const float* mc = mb + c * MH * MW;
      float g00 = mc[y0 * MW + x0];
      float g01 = mc[y0 * MW + x0 + 1];
      float g10 = mc[(y0 + 1) * MW + x0];
      float g11 = mc[(y0 + 1) * MW + x0 + 1];
      float top = g00 * (1.f - wx) + g01 * wx;
      float bot = g10 * (1.f - wx) + g11 * wx;
      v[c] = top * (1.f - wy) + bot * wy;
    }
    mx = v[0]; my = v[1]; mz = v[2];
    s = mx * mx + my * my + mz * mz;
  }
  float2* tb = tiles + ((size_t)b * NT + t) * 32;
  tb[nl]      = make_float2(mx, my);   // lanes 0..15 : (k0,k1)
  tb[16 + nl] = make_float2(mz, s);    // lanes 16..31: (k2,k3)
}

// ---------------------------------------------------------------------------
// WMMA step: one tile -> one v_wmma_f32_16x16x4_f32, fold into running min.
// ---------------------------------------------------------------------------
static __device__ __forceinline__ void wmma_min_step(const v2f& A, float2 bv,
                                                     float rmin[8]) {
  v2f Bf; Bf.x = bv.x; Bf.y = bv.y;
  v8f acc = {0.f, 0.f, 0.f, 0.f, 0.f, 0.f, 0.f, 0.f};
  acc = __builtin_amdgcn_wmma_f32_16x16x4_f32(
      /*neg_a=*/false, A, /*neg_b=*/false, Bf,
      /*c_mod=*/(short)0, acc, /*reuse_a=*/false, /*reuse_b=*/false);
#pragma unroll
  for (int r = 0; r < 8; ++r) rmin[r] = min_fast(rmin[r], acc[r]);
}

// ---------------------------------------------------------------------------
// Kernel 2: one wave32 per 16 pc points; 8 waves (one batch) per block.
// Wave 0 DMA-stages all 553 mesh tiles of this block's batch into LDS via
// the Tensor Data Mover, then every wave loops tiles from LDS:
//   ds_load_b64 (B fragment) -> v_wmma_f32_16x16x4_f32 -> v_min_num_f32 x8.
// EXEC is all-ones around the WMMAs (exact grid, no divergence in the loop).
// ---------------------------------------------------------------------------
__global__ void __launch_bounds__(256)
chamfer_min_kernel(const float* __restrict__ pc,
                   const float2* __restrict__ tiles,
                   float* __restrict__ partial) {
  extern __shared__ float2 smem[];               // NT*32 float2 = 141568 B

  const int lane = threadIdx.x & 31;
  const int wblk = threadIdx.x >> 5;             // wave id in block (0..7)
  const int wave = blockIdx.x * 8 + wblk;
  const int row  = lane & 15;
  const bool hi  = lane >= 16;
  const int b    = blockIdx.x >> 5;              // 32 blocks per batch
  const int mt   = wave & 255;                   // pc tile within batch
  const int m    = mt * 16 + row;

  // ---- TDM: DMA this batch's packed tiles (2D: 553 rows x 32 x 8B) to LDS.
  if (wblk == 0) {
    typedef __attribute__((address_space(3))) const void* lds_ptr_t;
    unsigned lds_addr = (unsigned)(size_t)(lds_ptr_t)&smem[0];
    unsigned long long ga =
        (unsigned long long)(const void*)(tiles + (size_t)b * NT * 32);
    // D# group 0: count=1 | lds_addr | global_addr[56:0] | type=2
    uint32x4 g0 = {1u, lds_addr, (unsigned)(ga & 0xffffffffu),
                   (unsigned)((ga >> 32) & 0x1ffffffu) | (2u << 30)};
    // D# group 1: data_size=3 (8B); tensor_dim0=32, tensor_dim1=553;
    //             tile_dim0=32, tile_dim1=553; tensor_dim0_stride=32.
    int32x8 g1 = {(int)(3u << 16),        // workgroup_mask=0, data_size=8B
                  (int)(32u << 16),       // tensor_dim0[15:0]
                  (int)(553u << 16),      // tensor_dim0[31:16]=0, tensor_dim1[15:0]
                  (int)(32u << 16),       // tensor_dim1[31:16]=0, tile_dim0
                  553,                    // tile_dim1 (tile_dim2=0)
                  32,                     // tensor_dim0_stride[31:0]
                  0, 0};                  // stride hi, tensor_dim1_stride
    int32x4 g2 = {0, 0, 0, 0};            // unused (2D tensor)
    int32x4 g3 = {0, 0, 0, 0};            // unused (2D tensor)
    int32x8 g4 = {0, 0, 0, 0, 0, 0, 0, 0};// trailing operand, zero-filled
    // clang-23 / amdgpu-toolchain 6-arg form:
    __builtin_amdgcn_tensor_load_to_lds(g0, g1, g2, g3, g4, /*cpol=*/0);
    __builtin_amdgcn_s_wait_tensorcnt(0);
  }
  __syncthreads();

  // ---- A fragment (ISA 7.12.2, 32-bit A 16x4): lane L<16 holds row L,
  // K=0 (v0) / K=1 (v1); lane L>=16 holds row L-16, K=2 (v0) / K=3 (v1).
  const float* pb = pc + (size_t)b * 3 * MPTS;
  const float px = pb[m];
  const float py = pb[MPTS + m];
  const float pz = pb[2 * MPTS + m];
  const float pn = px * px + py * py + pz * pz;

  v2f A;
  A.x = hi ? (-2.f * pz) : (-2.f * px);   // K=2 : K=0
  A.y = hi ? 1.0f        : (-2.f * py);   // K=3 : K=1

  float rmin[8];
#pragma unroll
  for (int r = 0; r < 8; ++r) rmin[r] = 3.0e38f;

  // ---- Main loop: batch 4 ds_load_b64 ahead of 4 WMMAs (553 = 138*4 + 1).
  int t = 0;
  for (; t + 4 <= NT; t += 4) {
    float2 b0 = smem[(t + 0) * 32 + lane];
    float2 b1 = smem[(t + 1) * 32 + lane];
    float2 b2 = smem[(t + 2) * 32 + lane];
    float2 b3 = smem[(t + 3) * 32 + lane];
    wmma_min_step(A, b0, rmin);
    wmma_min_step(A, b1, rmin);
    wmma_min_step(A, b2, rmin);
    wmma_min_step(A, b3, rmin);
  }
  for (; t < NT; ++t) wmma_min_step(A, smem[t * 32 + lane], rmin);

  // ---- Min over the 16 columns: D-fragment rows live per 16-lane half
  // (lanes 0-15: rows 0-7; lanes 16-31: rows 8-15), xor masks 1..8 suffice.
#pragma unroll
  for (int mask = 1; mask <= 8; mask <<= 1) {
#pragma unroll
    for (int r = 0; r < 8; ++r)
      rmin[r] = min_fast(rmin[r], __shfl_xor(rmin[r], mask, 32));
  }

  // ---- Wave partial: lane0 adds row-mins 0..7, lane16 adds rows 8..15,
  // lanes 0..15 contribute one ||p||^2 each (one per row).
  float contrib = hi ? 0.f : pn;
  if (row == 0) {
    float s = 0.f;
#pragma unroll
    for (int r = 0; r < 8; ++r) s += rmin[r];
    contrib += s;
  }
#pragma unroll
  for (int mask = 16; mask >= 1; mask >>= 1)
    contrib += __shfl_xor(contrib, mask, 32);

  if (lane == 0) partial[wave] = contrib;
}

// ---------------------------------------------------------------------------
// Kernel 3: deterministic final reduction of the 1024 wave partials.
// ---------------------------------------------------------------------------
__global__ void reduce_mean_kernel(const float* __restrict__ partial,
                                   float* __restrict__ out) {
  __shared__ float sh[256];
  int i = threadIdx.x;
  float v = partial[i] + partial[i + 256] + partial[i + 512] + partial[i + 768];
  sh[i] = v;
  __syncthreads();
  for (int off = 128; off > 0; off >>= 1) {
    if (i < off) sh[i] += sh[i + off];
    __syncthreads();
  }
  if (i == 0) out[0] = sh[0] * (1.0f / (float)(BATCH * MPTS));
}

// ---------------------------------------------------------------------------
extern "C" void kernel_launch(void* const* d_in, const int* in_sizes, int n_in,
                              void* d_out, int out_size, void* d_ws, size_t ws_size,
                              hipStream_t stream) {
  const float* mesh = (const float*)d_in[0];   // (4,3,32,32) f32
  const float* pc   = (const float*)d_in[1];   // (4,3,4096)  f32
  float* out        = (float*)d_out;           // scalar f32

  float*  partial = (float*)d_ws;                              // 1024 floats
  float2* tiles   = (float2*)((char*)d_ws + WS_TILES_OFF);     // 4*553*32 float2

  int n1 = BATCH * NT * 16;  // 35392 packer threads
  refine_pack_kernel<<<(n1 + 255) / 256, 256, 0, stream>>>(mesh, tiles);

  chamfer_min_kernel<<<(NWAVE * 32) / 256, 256, LDS_BYTES, stream>>>(pc, tiles,
                                                                     partial);

  reduce_mean_kernel<<<1, 256, 0, stream>>>(partial, out);
}